// EncoderLayer_27986006901273
// MI455X (gfx1250) — compile-verified
//
#include <hip/hip_runtime.h>
#include <math.h>

// AllSetTransformer encoder layer for MI455X (gfx1250, wave32).
// fp32 GEMMs use V_WMMA_F32_16X16X4_F32 with 16x32 output per wave;
// weight panels are staged LDS-side via global_load_async_to_lds_b128
// (ASYNCcnt). Segment softmax uses L2-resident f32 atomics.

#define HID    256
#define NHEADS 8

typedef float v2f __attribute__((ext_vector_type(2)));
typedef float v8f __attribute__((ext_vector_type(8)));

enum PIdx { IWk = 0, IBk, IWv, IBv, IAtt, IWff, IBff, IG0, IB0, IG1, IB1 };

__device__ __forceinline__ float warp_sum(float v) {
#pragma unroll
  for (int o = 16; o > 0; o >>= 1) v += __shfl_down(v, o, 32);
  return v;
}

// ---------------- WMMA fp32 GEMM: C[M x 256] = A[M x 256] @ W[256 x 256] (+bias) (+=C) ----
// Each wave owns a 16x32 output tile (two accumulators share one A fragment);
// 4 waves per WG share two 256x16 W panels staged in LDS by async copies.
__global__ __launch_bounds__(128) void k_gemm256(
    const float* __restrict__ A, const float* __restrict__ W,
    const float* __restrict__ bias, float* __restrict__ C, int M, int acc) {
  __shared__ float Wp0[HID * 16];  // columns n0 .. n0+15   (stride-16: bank-conflict-free)
  __shared__ float Wp1[HID * 16];  // columns n0+16 .. n0+31
  const int n0 = blockIdx.x * 32;

  // Async cache->LDS fill: one 16B chunk per lane per issue, ASYNCcnt-tracked.
  for (int chunk = threadIdx.x; chunk < HID * 4; chunk += 128) {
    int k = chunk >> 2, c4 = (chunk & 3) * 4;  // row k, 4-float sub-chunk
    unsigned l0 = (unsigned)(size_t)&Wp0[k * 16 + c4];
    unsigned l1 = (unsigned)(size_t)&Wp1[k * 16 + c4];
    unsigned long long g0 = (unsigned long long)(W + (size_t)k * HID + n0 + c4);
    unsigned long long g1 = g0 + 16 * sizeof(float);
    asm volatile("global_load_async_to_lds_b128 %0, %1, off"
                 :: "v"(l0), "v"(g0) : "memory");
    asm volatile("global_load_async_to_lds_b128 %0, %1, off"
                 :: "v"(l1), "v"(g1) : "memory");
  }
  asm volatile("s_wait_asynccnt 0x0" ::: "memory");
  __syncthreads();

  const int wave = threadIdx.x >> 5;
  const int lane = threadIdx.x & 31;
  const int mtile = blockIdx.y * 4 + wave;
  if (mtile * 16 >= M) return;          // wave-uniform exit: EXEC all-ones for WMMA
  const int m0  = mtile * 16;
  const int n   = lane & 15;
  const int kb  = (lane >> 4) * 2;      // A/B frag K-pair select (ISA 16x4 f32 layout)
  const int mhi = (lane >> 4) * 8;      // C/D frag M-half select
  v8f c0, c1;
  if (acc) {
#pragma unroll
    for (int j = 0; j < 8; ++j) {
      c0[j] = C[(size_t)(m0 + mhi + j) * HID + n0 + n];
      c1[j] = C[(size_t)(m0 + mhi + j) * HID + n0 + 16 + n];
    }
  } else {
#pragma unroll
    for (int j = 0; j < 8; ++j) { c0[j] = 0.0f; c1[j] = 0.0f; }
  }
  const float* Arow = A + (size_t)(m0 + (lane & 15)) * HID;
#pragma unroll 4
  for (int k = 0; k < HID; k += 4) {
    v2f a, b0, b1;
    a.x = Arow[k + kb];
    a.y = Arow[k + kb + 1];
    b0.x = Wp0[(k + kb) * 16 + n];
    b0.y = Wp0[(k + kb + 1) * 16 + n];
    b1.x = Wp1[(k + kb) * 16 + n];
    b1.y = Wp1[(k + kb + 1) * 16 + n];
    c0 = __builtin_amdgcn_wmma_f32_16x16x4_f32(false, a, false, b0, (short)0,
                                               c0, false, false);
    c1 = __builtin_amdgcn_wmma_f32_16x16x4_f32(false, a, false, b1, (short)0,
                                               c1, false, false);
  }
  if (bias) {
    float bv0 = bias[n0 + n];
    float bv1 = bias[n0 + 16 + n];
#pragma unroll
    for (int j = 0; j < 8; ++j) { c0[j] += bv0; c1[j] += bv1; }
  }
#pragma unroll
  for (int j = 0; j < 8; ++j) {
    C[(size_t)(m0 + mhi + j) * HID + n0 + n]      = c0[j];
    C[(size_t)(m0 + mhi + j) * HID + n0 + 16 + n] = c1[j];
  }
}

// ---------------- alpha = x @ (Wk . att) folding --------------------------------------
__global__ void k_prep_att(const float* __restrict__ Wk, const float* __restrict__ bk,
                           const float* __restrict__ att, float* __restrict__ wkatt,
                           float* __restrict__ ab) {
  int c = threadIdx.x;  // 256 threads
#pragma unroll
  for (int h = 0; h < NHEADS; ++h) {
    float s = 0.f;
    for (int d = 0; d < 32; ++d) s += Wk[c * HID + h * 32 + d] * att[h * 32 + d];
    wkatt[c * NHEADS + h] = s;
  }
  if (c < NHEADS) {
    float s = 0.f;
    for (int d = 0; d < 32; ++d) s += bk[c * 32 + d] * att[c * 32 + d];
    ab[c] = s;
  }
}

__global__ __launch_bounds__(256) void k_alpha(
    const float* __restrict__ x, const float* __restrict__ wkatt,
    const float* __restrict__ ab, float* __restrict__ alpha, int M) {
  __shared__ float sw[HID * NHEADS];
  __shared__ float sab[NHEADS];
  for (int i = threadIdx.x; i < HID * NHEADS; i += 256) sw[i] = wkatt[i];
  if (threadIdx.x < NHEADS) sab[threadIdx.x] = ab[threadIdx.x];
  __syncthreads();
  int wave = threadIdx.x >> 5, lane = threadIdx.x & 31;
  int node = blockIdx.x * 8 + wave;
  if (node >= M) return;
  float acc[NHEADS];
#pragma unroll
  for (int h = 0; h < NHEADS; ++h) acc[h] = 0.f;
  const float* xr = x + (size_t)node * HID;
#pragma unroll
  for (int j = 0; j < 8; ++j) {
    int c = j * 32 + lane;
    float xv = xr[c];
#pragma unroll
    for (int h = 0; h < NHEADS; ++h) acc[h] += xv * sw[c * NHEADS + h];
  }
#pragma unroll
  for (int h = 0; h < NHEADS; ++h) {
    float r = warp_sum(acc[h]);
    if (lane == 0) alpha[(size_t)node * NHEADS + h] = r + sab[h];
  }
}

// ---------------- segment softmax over incidence edges --------------------------------
__device__ __forceinline__ unsigned enc_f(float f) {
  unsigned u = __float_as_uint(f);
  return (u & 0x80000000u) ? ~u : (u | 0x80000000u);
}
__device__ __forceinline__ float dec_f(unsigned u) {
  return (u & 0x80000000u) ? __uint_as_float(u & 0x7fffffffu)
                           : __uint_as_float(~u);
}

__global__ void k_fill_u32(unsigned* __restrict__ p, unsigned v, int n) {
  int i = blockIdx.x * blockDim.x + threadIdx.x;
  if (i < n) p[i] = v;
}

__global__ void k_edge_max(const float* __restrict__ alpha, const int* __restrict__ src,
                           const int* __restrict__ dst, unsigned* __restrict__ amax,
                           int nE) {
  int i = blockIdx.x * blockDim.x + threadIdx.x;
  if (i >= nE * NHEADS) return;
  int e = i >> 3, h = i & 7;
  float a = alpha[(size_t)src[e] * NHEADS + h];
  a = a > 0.f ? a : 0.2f * a;  // leaky_relu(0.2)
  atomicMax(&amax[(size_t)dst[e] * NHEADS + h], enc_f(a));
}

__global__ void k_amax_fix(unsigned* __restrict__ amax, int n) {
  int i = blockIdx.x * blockDim.x + threadIdx.x;
  if (i >= n) return;
  float f = dec_f(amax[i]);
  if (!isfinite(f)) f = 0.f;  // empty-segment guard, as in reference
  ((float*)amax)[i] = f;
}

__global__ void k_edge_exp(const float* __restrict__ alpha, const int* __restrict__ src,
                           const int* __restrict__ dst, const float* __restrict__ amax,
                           float* __restrict__ ex, float* __restrict__ den, int nE) {
  int i = blockIdx.x * blockDim.x + threadIdx.x;
  if (i >= nE * NHEADS) return;
  int e = i >> 3, h = i & 7;
  float a = alpha[(size_t)src[e] * NHEADS + h];
  a = a > 0.f ? a : 0.2f * a;
  float v = __expf(a - amax[(size_t)dst[e] * NHEADS + h]);
  ex[i] = v;
  __hip_atomic_fetch_add(&den[(size_t)dst[e] * NHEADS + h], v, __ATOMIC_RELAXED,
                         __HIP_MEMORY_SCOPE_AGENT);
}

// one wave per edge: 8 channels per lane, f32 atomic adds resolve in L2
__global__ __launch_bounds__(256) void k_edge_scatter(
    const float* __restrict__ xV, const int* __restrict__ src,
    const int* __restrict__ dst, const float* __restrict__ ex,
    const float* __restrict__ den, float* __restrict__ out, int nE) {
  int wave = (int)((blockIdx.x * blockDim.x + threadIdx.x) >> 5);
  int lane = threadIdx.x & 31;
  if (wave >= nE) return;
  int s = src[wave], d = dst[wave];
  const float* xr = xV + (size_t)s * HID;
  float* orow = out + (size_t)d * HID;
#pragma unroll
  for (int j = 0; j < NHEADS; ++j) {
    float w = ex[(size_t)wave * NHEADS + j] /
              (den[(size_t)d * NHEADS + j] + 1e-16f);
    int c = j * 32 + lane;
    __hip_atomic_fetch_add(&orow[c], xr[c] * w, __ATOMIC_RELAXED,
                           __HIP_MEMORY_SCOPE_AGENT);
  }
}

// ---------------- rowwise LayerNorm epilogues -----------------------------------------
__device__ __forceinline__ float block_ln(float v, float* sred) {
  int lane = threadIdx.x & 31, wv = threadIdx.x >> 5;
  float s = warp_sum(v);
  if (lane == 0) sred[wv] = s;
  __syncthreads();
  if (wv == 0) {
    float t = (lane < 8) ? sred[lane] : 0.f;
    t = warp_sum(t);
    if (lane == 0) sred[0] = t;
  }
  __syncthreads();
  float mean = sred[0] * (1.0f / HID);
  __syncthreads();
  float dv = v - mean;
  float s2 = warp_sum(dv * dv);
  if (lane == 0) sred[wv] = s2;
  __syncthreads();
  if (wv == 0) {
    float t = (lane < 8) ? sred[lane] : 0.f;
    t = warp_sum(t);
    if (lane == 0) sred[0] = t;
  }
  __syncthreads();
  float var = sred[0] * (1.0f / HID);
  float r = dv * rsqrtf(var + 1e-5f);
  __syncthreads();  // make sred reusable
  return r;
}

__global__ __launch_bounds__(256) void k_row_pre(
    const float* __restrict__ pooled, const float* __restrict__ att,
    const float* __restrict__ instr, const float* __restrict__ g,
    const float* __restrict__ b, float* __restrict__ y, int M) {
  __shared__ float sred[8];
  int r = blockIdx.x;
  if (r >= M) return;
  int c = threadIdx.x;
  float v = pooled[(size_t)r * HID + c] + att[c];  // seed residual
  if (instr) v += instr[(size_t)r * HID + c];
  float nrm = block_ln(v, sred);
  y[(size_t)r * HID + c] = nrm * g[c] + b[c];
}

__global__ __launch_bounds__(256) void k_row_post(
    const float* __restrict__ y, const float* __restrict__ f,
    const float* __restrict__ g, const float* __restrict__ b,
    float* __restrict__ out, int M) {
  __shared__ float sred[8];
  int r = blockIdx.x;
  if (r >= M) return;
  int c = threadIdx.x;
  float fv = f[(size_t)r * HID + c];
  float v = y[(size_t)r * HID + c] + (fv > 0.f ? fv : 0.f);  // rFF residual
  float nrm = block_ln(v, sred);
  float o = nrm * g[c] + b[c];
  out[(size_t)r * HID + c] = o > 0.f ? o : 0.f;  // caller's relu
}

// ---------------- host-side stage driver ----------------------------------------------
static void run_allset(const float* x, int Msrc, const int* src, const int* dst,
                       int Mdst, const float* const* P, const float* instr,
                       float* result, float* wkatt, float* ab, float* alpha,
                       float* amax, float* den, float* exb, float* xVf,
                       float* pooled, float* ybuf, int nE, hipStream_t stream) {
  k_prep_att<<<1, 256, 0, stream>>>(P[IWk], P[IBk], P[IAtt], wkatt, ab);
  k_alpha<<<(Msrc + 7) / 8, 256, 0, stream>>>(x, wkatt, ab, alpha, Msrc);
  dim3 gg(8, (unsigned)((Msrc / 16 + 3) / 4));
  k_gemm256<<<gg, 128, 0, stream>>>(x, P[IWv], P[IBv], xVf, Msrc, 0);
  hipMemsetAsync(den, 0, (size_t)Mdst * NHEADS * sizeof(float), stream);
  hipMemsetAsync(pooled, 0, (size_t)Mdst * HID * sizeof(float), stream);
  k_fill_u32<<<(Mdst * NHEADS + 255) / 256, 256, 0, stream>>>(
      (unsigned*)amax, 0x007fffffu /* enc(-inf) */, Mdst * NHEADS);
  k_edge_max<<<(nE * NHEADS + 255) / 256, 256, 0, stream>>>(alpha, src, dst,
                                                            (unsigned*)amax, nE);
  k_amax_fix<<<(Mdst * NHEADS + 255) / 256, 256, 0, stream>>>((unsigned*)amax,
                                                              Mdst * NHEADS);
  k_edge_exp<<<(nE * NHEADS + 255) / 256, 256, 0, stream>>>(alpha, src, dst,
                                                            amax, exb, den, nE);
  k_edge_scatter<<<(nE + 7) / 8, 256, 0, stream>>>(xVf, src, dst, exb, den,
                                                   pooled, nE);
  k_row_pre<<<Mdst, 256, 0, stream>>>(pooled, P[IAtt], instr, P[IG0], P[IB0],
                                      ybuf, Mdst);
  dim3 gf(8, (unsigned)((Mdst / 16 + 3) / 4));
  k_gemm256<<<gf, 128, 0, stream>>>(ybuf, P[IWff], P[IBff], xVf, Mdst, 0);
  k_row_post<<<Mdst, 256, 0, stream>>>(ybuf, xVf, P[IG1], P[IB1], result, Mdst);
}

extern "C" void kernel_launch(void* const* d_in, const int* in_sizes, int n_in,
                              void* d_out, int out_size, void* d_ws,
                              size_t ws_size, hipStream_t stream) {
  const float* emb_s = (const float*)d_in[0];
  const float* emb_t = (const float*)d_in[1];
  const float* emb_i = (const float*)d_in[2];
  const float* P[2][11];
  for (int s = 0; s < 2; ++s)
    for (int j = 0; j < 11; ++j) P[s][j] = (const float*)d_in[3 + s * 11 + j];
  const float* fuse_W = (const float*)d_in[25];
  const float* fuse_b = (const float*)d_in[26];
  const int* e_src = (const int*)d_in[27];
  const int* e_dst = (const int*)d_in[28];

  const int nN = in_sizes[0] / HID;   // 50000
  const int nH = in_sizes[1] / HID;   // 10000
  const int nE = in_sizes[27];        // 300000
  const int nMax = nN > nH ? nN : nH;

  float* s_out = (float*)d_out;                    // [nN, HID]
  float* t_out = s_out + (size_t)nN * HID;         // [nH, HID]

  // bump-carve workspace (lifetime-based aliasing keeps peak ~170 MB)
  float* w = (float*)d_ws;
  size_t off = 0;
  auto alloc = [&](size_t n) { float* r = w + off; off += n; return r; };
  size_t exSz = (size_t)nE * NHEADS;
  size_t ttSz = (size_t)nH * HID;
  float* wkatt  = alloc(HID * NHEADS);
  float* ab     = alloc(NHEADS);
  float* alpha  = alloc((size_t)nMax * NHEADS);
  float* amax   = alloc((size_t)nMax * NHEADS);
  float* den    = alloc((size_t)nMax * NHEADS);
  float* exb    = alloc(exSz > ttSz ? exSz : ttSz);  // ex (edge pass) ∪ t_tem
  float* xVf    = alloc((size_t)nMax * HID);         // xV (pre-scatter) ∪ FF out
  float* pooled = alloc((size_t)nMax * HID);
  float* ybuf   = alloc((size_t)nMax * HID);
  float* t_tem  = exb;  // alive only between row_post(V2E) and fuse2

  // ---- V2E: vertices -> hyperedges ----
  run_allset(emb_s, nN, e_src, e_dst, nH, P[0], nullptr, t_tem, wkatt, ab,
             alpha, amax, den, exb, xVf, pooled, ybuf, nE, stream);

  // ---- fuse: t = [emb_t | t_tem] @ fuse_W + fuse_b  (K=512 as two K=256 passes) ----
  dim3 gt(8, (unsigned)((nH / 16 + 3) / 4));
  k_gemm256<<<gt, 128, 0, stream>>>(emb_t, fuse_W, fuse_b, t_out, nH, 0);
  k_gemm256<<<gt, 128, 0, stream>>>(t_tem, fuse_W + (size_t)HID * HID, nullptr,
                                    t_out, nH, 1);

  // ---- E2V: hyperedges -> vertices (reversed incidence), + instruct ----
  run_allset(t_out, nH, e_dst, e_src, nN, P[1], emb_i, s_out, wkatt, ab, alpha,
             amax, den, exb, xVf, pooled, ybuf, nE, stream);
}